// AttentionFlow_23819888623966
// MI455X (gfx1250) — compile-verified
//
#include <hip/hip_runtime.h>

typedef float v2f __attribute__((ext_vector_type(2)));
typedef float v8f __attribute__((ext_vector_type(8)));

#define WPB 4  // waves per block in WMMA tile kernels (wave32)

__device__ __forceinline__ float leakyf(float x) { return x > 0.f ? x : 0.01f * x; }

// D = A(16x4) * B(4x16) + C, all f32 (exact f32 math, CDNA5 V_WMMA_F32_16X16X4_F32)
__device__ __forceinline__ v8f wmma4(v2f a, v2f b, v8f c) {
  return __builtin_amdgcn_wmma_f32_16x16x4_f32(false, a, false, b, (short)0, c, false, false);
}

// order-preserving float->u32 map (for atomicMax over possibly-negative logits)
__device__ __forceinline__ unsigned ordf(float f) {
  unsigned u = __float_as_uint(f);
  return (u & 0x80000000u) ? ~u : (u | 0x80000000u);
}
__device__ __forceinline__ float unordf(unsigned u) {
  unsigned b = (u & 0x80000000u) ? (u & 0x7FFFFFFFu) : ~u;
  return __uint_as_float(b);
}

// ---------------------------------------------------------------------------
// init scratch / output accumulators
__global__ void zero_kernel(unsigned* __restrict__ segmax, float* __restrict__ denom,
                            int* __restrict__ tieCnt, float* __restrict__ outScore,
                            int N, int B) {
  int i = blockIdx.x * blockDim.x + threadIdx.x;
  if (i < N) { segmax[i] = 0u; denom[i] = 0.f; outScore[i] = 0.f; }
  if (i < B) tieCnt[i] = 0;
}

// ---------------------------------------------------------------------------
// Pack row-major W[Ntot][K...] (row stride, column offset) into WMMA-B lane order:
// P[((s*numC + c)*32 + lane)*2 + j] = W[n][k],  k = s*4 + 2*(lane>>4) + j, n = c*16 + (lane&15)
__global__ void pack_kernel(const float* __restrict__ W, float* __restrict__ P,
                            int Ktot, int Ntot, int rowStride, int colOff) {
  int i = blockIdx.x * blockDim.x + threadIdx.x;
  int total = Ktot * Ntot;
  if (i >= total) return;
  int j = i & 1;
  int t = i >> 1;
  int lane = t & 31; t >>= 5;
  int numC = Ntot >> 4;
  int c = t % numC;
  int s = t / numC;
  int k = s * 4 + 2 * (lane >> 4) + j;
  int n = c * 16 + (lane & 15);
  P[i] = W[(size_t)n * rowStride + colOff + k];
}

// ---------------------------------------------------------------------------
// qL[b] = bl + qs(b)@Wl[:,128:160].T + qr(b)@Wl[:,160:224].T   (qR same with Wr/br)
// qs = q_src_ts@W_static.T + b_static ; qr = q_rel@W_proj.T + b_proj
__global__ __launch_bounds__(64) void q_kernel(
    const float* __restrict__ q_src_ts, const float* __restrict__ q_rel,
    const float* __restrict__ W_static, const float* __restrict__ b_static,
    const float* __restrict__ W_proj, const float* __restrict__ b_proj,
    const float* __restrict__ Wl, const float* __restrict__ bl,
    const float* __restrict__ Wr, const float* __restrict__ br,
    float* __restrict__ qL, float* __restrict__ qR) {
  int b = blockIdx.x;
  int t = threadIdx.x;  // 64 threads
  __shared__ float qs[32];
  __shared__ float qr[64];
  if (t < 32) {
    float a = b_static[t];
    const float* w = W_static + (size_t)t * 128;
    const float* x = q_src_ts + (size_t)b * 128;
    for (int j = 0; j < 128; ++j) a += w[j] * x[j];
    qs[t] = a;
  }
  {
    float a = b_proj[t];
    const float* w = W_proj + (size_t)t * 256;
    const float* x = q_rel + (size_t)b * 256;
    for (int j = 0; j < 256; ++j) a += w[j] * x[j];
    qr[t] = a;
  }
  __syncthreads();
  float accl = bl[t], accr = br[t];
  const float* wl = Wl + (size_t)t * 224;
  const float* wr = Wr + (size_t)t * 224;
  for (int j = 0; j < 32; ++j) { accl += qs[j] * wl[128 + j]; accr += qs[j] * wr[128 + j]; }
  for (int j = 0; j < 64; ++j) { accl += qr[j] * wl[160 + j]; accr += qr[j] * wr[160 + j]; }
  qL[(size_t)b * 64 + t] = accl;
  qR[(size_t)b * 64 + t] = accr;
}

// ---------------------------------------------------------------------------
// nodeL/nodeR: proj = vis@W_proj.T + b_proj (K=256), then @Wl[:,0:64].T and @Wr[:,0:64].T
__global__ __launch_bounds__(32 * WPB) void node_kernel(
    const float* __restrict__ vis, const float* __restrict__ Pproj,
    const float* __restrict__ b_proj,
    const float* __restrict__ PlN, const float* __restrict__ PrN,
    float* __restrict__ nodeL, float* __restrict__ nodeR, int N, int nTiles) {
  __shared__ float stage[WPB][16 * 64];
  int wave = threadIdx.x >> 5, lane = threadIdx.x & 31;
  int tile = blockIdx.x * WPB + wave;
  if (tile >= nTiles) return;
  float* sm = stage[wave];
  int l16 = lane & 15, half = lane >> 4;
  int row0 = tile * 16;
  int arow = row0 + l16; if (arow >= N) arow = N - 1;
  const float* aptr = vis + (size_t)arow * 256 + 2 * half;

  v8f acc[4];
#pragma unroll
  for (int c = 0; c < 4; ++c) {
    float bias = b_proj[c * 16 + l16];
#pragma unroll
    for (int i = 0; i < 8; ++i) acc[c][i] = bias;
  }
  for (int s = 0; s < 64; ++s) {
    v2f a = *(const v2f*)(aptr + s * 4);
    const float* bb = Pproj + (size_t)s * 256;
#pragma unroll
    for (int c = 0; c < 4; ++c) {
      v2f bv = *(const v2f*)(bb + c * 64 + lane * 2);
      acc[c] = wmma4(a, bv, acc[c]);
    }
  }
#pragma unroll
  for (int c = 0; c < 4; ++c)
#pragma unroll
    for (int v = 0; v < 8; ++v)
      sm[(v + 8 * half) * 64 + c * 16 + l16] = acc[c][v];

  v8f aL[4], aR[4];
#pragma unroll
  for (int c = 0; c < 4; ++c)
#pragma unroll
    for (int i = 0; i < 8; ++i) { aL[c][i] = 0.f; aR[c][i] = 0.f; }
  for (int s = 0; s < 16; ++s) {
    v2f a = *(const v2f*)(sm + l16 * 64 + s * 4 + 2 * half);
    const float* b1 = PlN + (size_t)s * 256;
    const float* b2 = PrN + (size_t)s * 256;
#pragma unroll
    for (int c = 0; c < 4; ++c) {
      v2f bv1 = *(const v2f*)(b1 + c * 64 + lane * 2);
      aL[c] = wmma4(a, bv1, aL[c]);
      v2f bv2 = *(const v2f*)(b2 + c * 64 + lane * 2);
      aR[c] = wmma4(a, bv2, aR[c]);
    }
  }
#pragma unroll
  for (int v = 0; v < 8; ++v) {
    int row = row0 + v + 8 * half;
    if (row < N) {
#pragma unroll
      for (int c = 0; c < 4; ++c) {
        nodeL[(size_t)row * 64 + c * 16 + l16] = aL[c][v];
        nodeR[(size_t)row * 64 + c * 16 + l16] = aR[c][v];
      }
    }
  }
}

// ---------------------------------------------------------------------------
// new_rep = leaky(vis @ W_step.T + b_step)   [N,256], K=256, 16 col-tiles (4 chunks of 4)
__global__ __launch_bounds__(32 * WPB) void step_kernel(
    const float* __restrict__ vis, const float* __restrict__ Pstep,
    const float* __restrict__ b_step, float* __restrict__ out, int N, int nTiles) {
  int wave = threadIdx.x >> 5, lane = threadIdx.x & 31;
  int tile = blockIdx.x * WPB + wave;
  if (tile >= nTiles) return;
  int l16 = lane & 15, half = lane >> 4;
  int row0 = tile * 16;
  int arow = row0 + l16; if (arow >= N) arow = N - 1;
  const float* aptr = vis + (size_t)arow * 256 + 2 * half;
  for (int cc = 0; cc < 4; ++cc) {
    v8f acc[4];
#pragma unroll
    for (int c = 0; c < 4; ++c) {
      float bias = b_step[cc * 64 + c * 16 + l16];
#pragma unroll
      for (int i = 0; i < 8; ++i) acc[c][i] = bias;
    }
    for (int s = 0; s < 64; ++s) {
      v2f a = *(const v2f*)(aptr + s * 4);
      const float* bb = Pstep + ((size_t)s * 16 + cc * 4) * 64;
#pragma unroll
      for (int c = 0; c < 4; ++c) {
        v2f bv = *(const v2f*)(bb + c * 64 + lane * 2);
        acc[c] = wmma4(a, bv, acc[c]);
      }
    }
#pragma unroll
    for (int v = 0; v < 8; ++v) {
      int row = row0 + v + 8 * half;
      if (row < N) {
#pragma unroll
        for (int c = 0; c < 4; ++c)
          out[(size_t)row * 256 + cc * 64 + c * 16 + l16] = leakyf(acc[c][v]);
      }
    }
  }
}

// ---------------------------------------------------------------------------
// Fused per-edge chain: relP -> relL/relR -> assemble L,R -> leaky -> M=leaky(R)@Wc.T+bc
// -> logits = rowsum(leaky(L)*M); then atomicMax segment max over src.
__global__ __launch_bounds__(32 * WPB) void edge_kernel(
    const float* __restrict__ rel_emb,
    const int* __restrict__ src_idx, const int* __restrict__ tgt_idx,
    const int* __restrict__ eg_idx,
    const float* __restrict__ nodeL, const float* __restrict__ nodeR,
    const float* __restrict__ qL, const float* __restrict__ qR,
    const float* __restrict__ Pproj, const float* __restrict__ PlR,
    const float* __restrict__ PrR, const float* __restrict__ Pc,
    const float* __restrict__ b_proj, const float* __restrict__ bc,
    float* __restrict__ logits, unsigned* __restrict__ segmax,
    int E, int nTiles) {
  __shared__ float stage[WPB][16 * 64];
  int wave = threadIdx.x >> 5, lane = threadIdx.x & 31;
  int tile = blockIdx.x * WPB + wave;
  if (tile >= nTiles) return;
  float* sm = stage[wave];
  int l16 = lane & 15, half = lane >> 4;
  int row0 = tile * 16;
  int arow = row0 + l16; if (arow >= E) arow = E - 1;
  const float* aptr = rel_emb + (size_t)arow * 256 + 2 * half;

  // GEMM1: relP[16x64] = rel_emb @ W_proj.T + b_proj  (K=256)
  v8f acc[4];
#pragma unroll
  for (int c = 0; c < 4; ++c) {
    float bias = b_proj[c * 16 + l16];
#pragma unroll
    for (int i = 0; i < 8; ++i) acc[c][i] = bias;
  }
  for (int s = 0; s < 64; ++s) {
    v2f a = *(const v2f*)(aptr + s * 4);
    const float* bb = Pproj + (size_t)s * 256;
#pragma unroll
    for (int c = 0; c < 4; ++c) {
      v2f bv = *(const v2f*)(bb + c * 64 + lane * 2);
      acc[c] = wmma4(a, bv, acc[c]);
    }
  }
  // stage relP (D-layout -> row major in this wave's LDS slice)
#pragma unroll
  for (int c = 0; c < 4; ++c)
#pragma unroll
    for (int v = 0; v < 8; ++v)
      sm[(v + 8 * half) * 64 + c * 16 + l16] = acc[c][v];

  // GEMM2: relL = relP@Wl[:,64:128].T , relR = relP@Wr[:,64:128].T  (K=64)
  v8f aL[4], aR[4];
#pragma unroll
  for (int c = 0; c < 4; ++c)
#pragma unroll
    for (int i = 0; i < 8; ++i) { aL[c][i] = 0.f; aR[c][i] = 0.f; }
  for (int s = 0; s < 16; ++s) {
    v2f a = *(const v2f*)(sm + l16 * 64 + s * 4 + 2 * half);
    const float* b1 = PlR + (size_t)s * 256;
    const float* b2 = PrR + (size_t)s * 256;
#pragma unroll
    for (int c = 0; c < 4; ++c) {
      v2f bv1 = *(const v2f*)(b1 + c * 64 + lane * 2);
      aL[c] = wmma4(a, bv1, aL[c]);
      v2f bv2 = *(const v2f*)(b2 + c * 64 + lane * 2);
      aR[c] = wmma4(a, bv2, aR[c]);
    }
  }

  // assemble L/R with node and query contributions (coalesced 64B row-segment gathers),
  // apply leaky; keep leaky(L) in aL, stage leaky(R) to LDS for the Wc GEMM.
#pragma unroll
  for (int v = 0; v < 8; ++v) {
    int m = v + 8 * half;
    int erow = row0 + m; if (erow >= E) erow = E - 1;
    int si = src_idx[erow];
    int ti = tgt_idx[erow];
    int gi = eg_idx[erow];
#pragma unroll
    for (int c = 0; c < 4; ++c) {
      int col = c * 16 + l16;
      float L = aL[c][v] + nodeL[(size_t)si * 64 + col] + qL[(size_t)gi * 64 + col];
      float R = aR[c][v] + nodeR[(size_t)ti * 64 + col] + qR[(size_t)gi * 64 + col];
      aL[c][v] = leakyf(L);
      sm[m * 64 + col] = leakyf(R);
    }
  }

  // GEMM3: M = leaky(R) @ Wc.T + bc  (K=64)
  v8f aM[4];
#pragma unroll
  for (int c = 0; c < 4; ++c) {
    float bias = bc[c * 16 + l16];
#pragma unroll
    for (int i = 0; i < 8; ++i) aM[c][i] = bias;
  }
  for (int s = 0; s < 16; ++s) {
    v2f a = *(const v2f*)(sm + l16 * 64 + s * 4 + 2 * half);
    const float* bb = Pc + (size_t)s * 256;
#pragma unroll
    for (int c = 0; c < 4; ++c) {
      v2f bv = *(const v2f*)(bb + c * 64 + lane * 2);
      aM[c] = wmma4(a, bv, aM[c]);
    }
  }

  // logits = rowsum(leaky(L) * M): partial per lane, then 16-lane half-wave reduction
#pragma unroll
  for (int v = 0; v < 8; ++v) {
    float sum = 0.f;
#pragma unroll
    for (int c = 0; c < 4; ++c) sum += aL[c][v] * aM[c][v];
    sum += __shfl_xor(sum, 1, 32);
    sum += __shfl_xor(sum, 2, 32);
    sum += __shfl_xor(sum, 4, 32);
    sum += __shfl_xor(sum, 8, 32);
    int erow = row0 + v + 8 * half;
    if (l16 == 0 && erow < E) {
      logits[erow] = sum;
      atomicMax(&segmax[src_idx[erow]], ordf(sum));
    }
  }
}

// ---------------------------------------------------------------------------
__global__ void e2_kernel(const float* __restrict__ logits, const int* __restrict__ src_idx,
                          const unsigned* __restrict__ segmax, float* __restrict__ denom,
                          float* __restrict__ expv, int E) {
  int e = blockIdx.x * blockDim.x + threadIdx.x;
  if (e >= E) return;
  int s = src_idx[e];
  float ev = expf(logits[e] - unordf(segmax[s]));
  expv[e] = ev;
  atomicAdd(&denom[s], ev);
}

__global__ void e3_kernel(const float* __restrict__ expv, const float* __restrict__ denom,
                          const float* __restrict__ node_score, const int* __restrict__ src_idx,
                          float* __restrict__ att, int E) {
  int e = blockIdx.x * blockDim.x + threadIdx.x;
  if (e >= E) return;
  int s = src_idx[e];
  att[e] = (expv[e] / denom[s]) * node_score[s];
}

// group start offsets via binary search (eg_idx sorted ascending)
__global__ void starts_kernel(const int* __restrict__ eg, int* __restrict__ starts, int E, int B) {
  int b = blockIdx.x * blockDim.x + threadIdx.x;
  if (b > B) return;
  int lo = 0, hi = E;
  while (lo < hi) {
    int mid = (lo + hi) >> 1;
    if (eg[mid] < b) lo = mid + 1; else hi = mid;
  }
  starts[b] = lo;
}

// per-group k-th largest threshold via 4-pass radix select (att >= 0 -> bits monotonic)
__global__ __launch_bounds__(256) void topk_kernel(const float* __restrict__ att,
                                                   const int* __restrict__ starts,
                                                   const int* __restrict__ kptr,
                                                   unsigned* __restrict__ Tbits,
                                                   int* __restrict__ tieAllow) {
  int b = blockIdx.x;
  int beg = starts[b], end = starts[b + 1];
  int len = end - beg;
  int k = kptr[0];
  __shared__ unsigned hist[256];
  __shared__ unsigned sPrefix;
  __shared__ int sRemaining;
  if (len <= k) {
    if (threadIdx.x == 0) { Tbits[b] = 0u; tieAllow[b] = k; }
    return;
  }
  if (threadIdx.x == 0) { sPrefix = 0u; sRemaining = k; }
  __syncthreads();
  for (int p = 3; p >= 0; --p) {
    for (int i = threadIdx.x; i < 256; i += blockDim.x) hist[i] = 0u;
    __syncthreads();
    unsigned prefix = sPrefix;
    for (int e = beg + threadIdx.x; e < end; e += blockDim.x) {
      unsigned u = __float_as_uint(att[e]);
      bool match = (p == 3) || ((u >> ((p + 1) * 8)) == (prefix >> ((p + 1) * 8)));
      if (match) atomicAdd(&hist[(u >> (p * 8)) & 255u], 1u);
    }
    __syncthreads();
    if (threadIdx.x == 0) {
      int rem = sRemaining;
      int cum = 0;
      for (int d = 255; d >= 0; --d) {
        int c = (int)hist[d];
        if (cum + c >= rem) {
          sPrefix = prefix | ((unsigned)d << (p * 8));
          sRemaining = rem - cum;
          break;
        }
        cum += c;
      }
    }
    __syncthreads();
  }
  if (threadIdx.x == 0) { Tbits[b] = sPrefix; tieAllow[b] = sRemaining; }
}

// keep mask, pruned_att output, scatter-add of new_node_score over tgt
__global__ void e4_kernel(const float* __restrict__ att, const int* __restrict__ eg_idx,
                          const int* __restrict__ tgt_idx, const unsigned* __restrict__ Tbits,
                          const int* __restrict__ tieAllow, int* __restrict__ tieCnt,
                          float* __restrict__ pruned, float* __restrict__ outScore, int E) {
  int e = blockIdx.x * blockDim.x + threadIdx.x;
  if (e >= E) return;
  int g = eg_idx[e];
  float a = att[e];
  unsigned u = __float_as_uint(a);
  unsigned T = Tbits[g];
  bool keep = u > T;
  if (u == T) {
    int r = atomicAdd(&tieCnt[g], 1);
    keep = r < tieAllow[g];
  }
  float pa = keep ? a : 0.f;
  pruned[e] = pa;
  if (keep) atomicAdd(&outScore[tgt_idx[e]], pa);
}

// ---------------------------------------------------------------------------
extern "C" void kernel_launch(void* const* d_in, const int* in_sizes, int n_in,
                              void* d_out, int out_size, void* d_ws, size_t ws_size,
                              hipStream_t stream) {
  const float* node_score = (const float*)d_in[0];
  const float* visited    = (const float*)d_in[1];
  const float* rel_emb    = (const float*)d_in[2];
  const float* q_src_ts   = (const float*)d_in[3];
  const float* q_rel      = (const float*)d_in[4];
  const int*   eg_idx     = (const int*)d_in[5];
  const int*   src_idx    = (const int*)d_in[6];
  const int*   tgt_idx    = (const int*)d_in[7];
  const int*   kptr       = (const int*)d_in[8];
  const float* W_proj     = (const float*)d_in[9];
  const float* b_proj     = (const float*)d_in[10];
  const float* W_static   = (const float*)d_in[11];
  const float* b_static   = (const float*)d_in[12];
  const float* Wl         = (const float*)d_in[13];
  const float* bl         = (const float*)d_in[14];
  const float* Wr         = (const float*)d_in[15];
  const float* br         = (const float*)d_in[16];
  const float* Wc         = (const float*)d_in[17];
  const float* bc         = (const float*)d_in[18];
  const float* W_step     = (const float*)d_in[19];
  const float* b_step     = (const float*)d_in[20];

  const int N = in_sizes[0];
  const int E = in_sizes[5];
  const int D = in_sizes[1] / N;   // 256
  const int B = in_sizes[4] / D;   // 128

  float* out = (float*)d_out;
  float* outScore  = out;                          // [N]
  float* outRep    = out + N;                      // [N,256]
  float* outPruned = out + N + (size_t)N * 256;    // [E]

  float* ws = (float*)d_ws;
  size_t off = 0;
  auto take = [&](size_t n) { float* p = ws + off; off += (n + 1) & ~(size_t)1; return p; };
  float*    nodeL  = take((size_t)N * 64);
  float*    nodeR  = take((size_t)N * 64);
  float*    qL     = take((size_t)B * 64);
  float*    qR     = take((size_t)B * 64);
  float*    logits = take(E);
  float*    expv   = take(E);
  float*    att    = take(E);
  unsigned* segmax = (unsigned*)take(N);
  float*    denom  = take(N);
  int*      starts = (int*)take(B + 1);
  unsigned* Tbits  = (unsigned*)take(B);
  int*      tieAl  = (int*)take(B);
  int*      tieCnt = (int*)take(B);
  float*    Pproj  = take(256 * 64);
  float*    PlN    = take(64 * 64);
  float*    PrN    = take(64 * 64);
  float*    PlR    = take(64 * 64);
  float*    PrR    = take(64 * 64);
  float*    Pc     = take(64 * 64);
  float*    Pstep  = take(256 * 256);
  (void)ws_size; (void)n_in; (void)out_size;

  zero_kernel<<<(N + 255) / 256, 256, 0, stream>>>(segmax, denom, tieCnt, outScore, N, B);

  pack_kernel<<<(256 * 64 + 255) / 256, 256, 0, stream>>>(W_proj, Pproj, 256, 64, 256, 0);
  pack_kernel<<<(64 * 64 + 255) / 256, 256, 0, stream>>>(Wl, PlN, 64, 64, 224, 0);
  pack_kernel<<<(64 * 64 + 255) / 256, 256, 0, stream>>>(Wr, PrN, 64, 64, 224, 0);
  pack_kernel<<<(64 * 64 + 255) / 256, 256, 0, stream>>>(Wl, PlR, 64, 64, 224, 64);
  pack_kernel<<<(64 * 64 + 255) / 256, 256, 0, stream>>>(Wr, PrR, 64, 64, 224, 64);
  pack_kernel<<<(64 * 64 + 255) / 256, 256, 0, stream>>>(Wc, Pc, 64, 64, 64, 0);
  pack_kernel<<<(256 * 256 + 255) / 256, 256, 0, stream>>>(W_step, Pstep, 256, 256, 256, 0);

  q_kernel<<<B, 64, 0, stream>>>(q_src_ts, q_rel, W_static, b_static, W_proj, b_proj,
                                 Wl, bl, Wr, br, qL, qR);

  const int nTilesN = (N + 15) / 16;
  node_kernel<<<(nTilesN + WPB - 1) / WPB, 32 * WPB, 0, stream>>>(
      visited, Pproj, b_proj, PlN, PrN, nodeL, nodeR, N, nTilesN);
  step_kernel<<<(nTilesN + WPB - 1) / WPB, 32 * WPB, 0, stream>>>(
      visited, Pstep, b_step, outRep, N, nTilesN);

  const int nTilesE = (E + 15) / 16;
  edge_kernel<<<(nTilesE + WPB - 1) / WPB, 32 * WPB, 0, stream>>>(
      rel_emb, src_idx, tgt_idx, eg_idx, nodeL, nodeR, qL, qR,
      Pproj, PlR, PrR, Pc, b_proj, bc, logits, segmax, E, nTilesE);

  e2_kernel<<<(E + 255) / 256, 256, 0, stream>>>(logits, src_idx, segmax, denom, expv, E);
  e3_kernel<<<(E + 255) / 256, 256, 0, stream>>>(expv, denom, node_score, src_idx, att, E);
  starts_kernel<<<1, 256, 0, stream>>>(eg_idx, starts, E, B);
  topk_kernel<<<B, 256, 0, stream>>>(att, starts, kptr, Tbits, tieAl);
  e4_kernel<<<(E + 255) / 256, 256, 0, stream>>>(att, eg_idx, tgt_idx, Tbits, tieAl, tieCnt,
                                                 outPruned, outScore, E);
}